// SpeakerVerificationHead_70265664962962
// MI455X (gfx1250) — compile-verified
//
#include <hip/hip_runtime.h>
#include <hip/hip_bf16.h>
#include <math.h>

// ---------------------------------------------------------------------------
// SpeakerVerificationHead for MI455X (gfx1250, wave32, WMMA)
//
//   h        = relu(features @ W1^T + b1)              [2048, 512]
//   emb      = h @ W2^T + b2                           [2048, 256]
//   emb_norm = l2norm(emb)                             [2048, 256]
//   wn_inv   = 1 / ||arc_weight[s]||                   [50000]      (folded)
//   logits   = arcmargin(emb_norm @ arc_weight^T * wn_inv) * 30     [2048, 50000]
//   probs    = softmax(logits)                         [2048, 50000]
// ---------------------------------------------------------------------------

typedef __attribute__((ext_vector_type(2))) float v2f;
typedef __attribute__((ext_vector_type(4))) float v4f;
typedef __attribute__((ext_vector_type(8))) float v8f;

#define COS_M 0.980066577841242f
#define SIN_M 0.198669330795061f
#define ARC_SCALE 30.0f

// 16 WMMAs covering one K16 chunk for 4 N-subtiles.
// K-slot remap (valid because the K reduction is order independent):
//   lanes 0-15  (half=0) supply memory K in [16g,    16g+8)
//   lanes 16-31 (half=1) supply memory K in [16g+8,  16g+16)
// a0/b0 hold floats [base .. base+3], a1/b1 hold [base+4 .. base+7],
// where base = 16g + 8*half.  WMMA step s uses element pair {2s, 2s+1}
// of that per-lane 8-float window, identically for A and B.
__device__ __forceinline__ void wmma_group(v8f (&acc)[4], v4f a0, v4f a1,
                                           const v4f (&b0)[4], const v4f (&b1)[4])
{
    v2f a;
    a = __builtin_shufflevector(a0, a0, 0, 1);
#pragma unroll
    for (int t = 0; t < 4; ++t) {
        v2f b = __builtin_shufflevector(b0[t], b0[t], 0, 1);
        acc[t] = __builtin_amdgcn_wmma_f32_16x16x4_f32(false, a, false, b, (short)0, acc[t], false, false);
    }
    a = __builtin_shufflevector(a0, a0, 2, 3);
#pragma unroll
    for (int t = 0; t < 4; ++t) {
        v2f b = __builtin_shufflevector(b0[t], b0[t], 2, 3);
        acc[t] = __builtin_amdgcn_wmma_f32_16x16x4_f32(false, a, false, b, (short)0, acc[t], false, false);
    }
    a = __builtin_shufflevector(a1, a1, 0, 1);
#pragma unroll
    for (int t = 0; t < 4; ++t) {
        v2f b = __builtin_shufflevector(b1[t], b1[t], 0, 1);
        acc[t] = __builtin_amdgcn_wmma_f32_16x16x4_f32(false, a, false, b, (short)0, acc[t], false, false);
    }
    a = __builtin_shufflevector(a1, a1, 2, 3);
#pragma unroll
    for (int t = 0; t < 4; ++t) {
        v2f b = __builtin_shufflevector(b1[t], b1[t], 2, 3);
        acc[t] = __builtin_amdgcn_wmma_f32_16x16x4_f32(false, a, false, b, (short)0, acc[t], false, false);
    }
}

// ---------------------------------------------------------------------------
// Templated NT GEMM:  C[M,N] = epilogue( A[M,K] * B[N,K]^T )
// A, B row-major, K contiguous.  fp32 WMMA 16x16x4, K%16 == 0.
// Block = 256 threads (8 waves). Block tile 128(M) x 64(N); all 8 waves share
// the same B fragments -> 8x WGP$ reuse of the row-scattered B loads.
// Wave tile = 16(M) x 64(N): 4 accumulators, A-fragment reused 4x.
// One-chunk software pipeline: chunk g+1 fragments prefetched (b128 loads)
// before chunk g's 16 WMMAs issue.
// MODE 0: relu(x + bias[col])          (GEMM1)
// MODE 1: x + bias[col]                (GEMM2)
// MODE 2: ArcMargin epilogue -> logits (big GEMM)
// ---------------------------------------------------------------------------
template <int MODE>
__global__ __launch_bounds__(256) void gemm_nt_wmma(
    const float* __restrict__ A, const float* __restrict__ B,
    const float* __restrict__ bias, float* __restrict__ C,
    int M, int N, int K,
    const int* __restrict__ labels, const float* __restrict__ wn_inv)
{
    const int wave = threadIdx.x >> 5;   // 0..7
    const int lane = threadIdx.x & 31;
    const int half = lane >> 4;          // 0: lanes 0-15, 1: lanes 16-31
    const int l16  = lane & 15;

    const int row0 = blockIdx.y * 128 + wave * 16;  // M tile base for this wave
    const int col0 = blockIdx.x * 64;               // N tile base for this block

    // Per-lane 8-float K window base (see wmma_group comment).
    const float* __restrict__ arow = A + (size_t)(row0 + l16) * K + 8 * half;

    // B rows for the 4 N-subtiles (clamp OOB columns; results masked at store:
    // each output column depends only on its own B row).
    const float* __restrict__ brow[4];
#pragma unroll
    for (int t = 0; t < 4; ++t) {
        int c = col0 + t * 16 + l16;
        if (c > N - 1) c = N - 1;
        brow[t] = B + (size_t)c * K + 8 * half;
    }

    v8f acc[4];
#pragma unroll
    for (int t = 0; t < 4; ++t) acc[t] = (v8f)0.0f;

    const int ng = K >> 4;   // K16 chunks

    // Prologue: load chunk 0 fragments.
    v4f a0 = *(const v4f*)(arow);
    v4f a1 = *(const v4f*)(arow + 4);
    v4f b0[4], b1[4];
#pragma unroll
    for (int t = 0; t < 4; ++t) {
        b0[t] = *(const v4f*)(brow[t]);
        b1[t] = *(const v4f*)(brow[t] + 4);
    }

    // Pipelined main loop.
    for (int g = 0; g < ng - 1; ++g) {
        const int off = (g + 1) * 16;
        v4f na0 = *(const v4f*)(arow + off);
        v4f na1 = *(const v4f*)(arow + off + 4);
        v4f nb0[4], nb1[4];
#pragma unroll
        for (int t = 0; t < 4; ++t) {
            nb0[t] = *(const v4f*)(brow[t] + off);
            nb1[t] = *(const v4f*)(brow[t] + off + 4);
        }

        wmma_group(acc, a0, a1, b0, b1);

        a0 = na0; a1 = na1;
#pragma unroll
        for (int t = 0; t < 4; ++t) { b0[t] = nb0[t]; b1[t] = nb1[t]; }
    }
    // Drain last chunk.
    wmma_group(acc, a0, a1, b0, b1);

    // Epilogue + store.  C/D layout: lane -> column (l16), VGPR v -> row
    // (row0 + v + 8*half).
#pragma unroll
    for (int t = 0; t < 4; ++t) {
        const int col = col0 + t * 16 + l16;
        if (col >= N) continue;
#pragma unroll
        for (int v = 0; v < 8; ++v) {
            const int row = row0 + v + 8 * half;
            float x = acc[t][v];
            float val;
            if (MODE == 0) {
                val = fmaxf(x + bias[col], 0.0f);
            } else if (MODE == 1) {
                val = x + bias[col];
            } else {
                // cos(theta) via folded weight normalization
                float c = x * wn_inv[col];
                c = fminf(1.0f, fmaxf(-1.0f, c));
                float s  = sqrtf(fmaxf(1.0f - c * c, 0.0f));
                float cm = c * COS_M - s * SIN_M;
                val = ((labels[row] == col) ? cm : c) * ARC_SCALE;
            }
            C[(size_t)row * N + col] = val;
        }
    }
}

// ---------------------------------------------------------------------------
// emb_norm[row] = emb[row] / max(||emb[row]||, 1e-12)   (one block per row,
// 256 threads == 256 columns)
// ---------------------------------------------------------------------------
__global__ __launch_bounds__(256) void rownorm_kernel(
    const float* __restrict__ emb, float* __restrict__ emb_norm)
{
    __shared__ float red[8];
    const int row = blockIdx.x;
    const int tid = threadIdx.x;

    float x  = emb[(size_t)row * 256 + tid];
    float ss = x * x;
#pragma unroll
    for (int off = 16; off > 0; off >>= 1) ss += __shfl_xor(ss, off, 32);
    if ((tid & 31) == 0) red[tid >> 5] = ss;
    __syncthreads();
    if (tid == 0) {
        float s = 0.0f;
#pragma unroll
        for (int i = 0; i < 8; ++i) s += red[i];
        red[0] = 1.0f / fmaxf(sqrtf(s), 1e-12f);
    }
    __syncthreads();
    emb_norm[(size_t)row * 256 + tid] = x * red[0];
}

// ---------------------------------------------------------------------------
// wn_inv[row] = 1 / max(||arc_weight[row]||, 1e-12)   (one wave per row)
// ---------------------------------------------------------------------------
__global__ __launch_bounds__(256) void wnorm_kernel(
    const float* __restrict__ w, float* __restrict__ inv, int nrows)
{
    const int wave = threadIdx.x >> 5;
    const int lane = threadIdx.x & 31;
    const int row  = blockIdx.x * 8 + wave;
    if (row >= nrows) return;

    const float* __restrict__ p = w + (size_t)row * 256;
    float ss = 0.0f;
#pragma unroll
    for (int i = 0; i < 8; ++i) {
        float x = p[lane + 32 * i];
        ss += x * x;
    }
#pragma unroll
    for (int off = 16; off > 0; off >>= 1) ss += __shfl_xor(ss, off, 32);
    if (lane == 0) inv[row] = 1.0f / fmaxf(sqrtf(ss), 1e-12f);
}

// ---------------------------------------------------------------------------
// probs[row] = softmax(logits[row])  over N=50000.  One 256-thread block per
// row; the 200KB row stays L2-resident across the three passes.
// ---------------------------------------------------------------------------
__global__ __launch_bounds__(256) void softmax_kernel(
    const float* __restrict__ logits, float* __restrict__ probs, int N)
{
    __shared__ float red[8];
    __shared__ float bcast;
    const size_t base = (size_t)blockIdx.x * N;
    const int tid = threadIdx.x;

    // Pass 1: row max
    float m = -INFINITY;
    for (int i = tid; i < N; i += 256) m = fmaxf(m, logits[base + i]);
#pragma unroll
    for (int off = 16; off > 0; off >>= 1) m = fmaxf(m, __shfl_xor(m, off, 32));
    if ((tid & 31) == 0) red[tid >> 5] = m;
    __syncthreads();
    if (tid == 0) {
        float mm = red[0];
#pragma unroll
        for (int i = 1; i < 8; ++i) mm = fmaxf(mm, red[i]);
        bcast = mm;
    }
    __syncthreads();
    m = bcast;
    __syncthreads();

    // Pass 2: sum of exp
    float s = 0.0f;
    for (int i = tid; i < N; i += 256) s += __expf(logits[base + i] - m);
#pragma unroll
    for (int off = 16; off > 0; off >>= 1) s += __shfl_xor(s, off, 32);
    if ((tid & 31) == 0) red[tid >> 5] = s;
    __syncthreads();
    if (tid == 0) {
        float ss = 0.0f;
#pragma unroll
        for (int i = 0; i < 8; ++i) ss += red[i];
        bcast = 1.0f / ss;
    }
    __syncthreads();
    const float inv = bcast;

    // Pass 3: write probs
    for (int i = tid; i < N; i += 256)
        probs[base + i] = __expf(logits[base + i] - m) * inv;
}

// ---------------------------------------------------------------------------
extern "C" void kernel_launch(void* const* d_in, const int* in_sizes, int n_in,
                              void* d_out, int out_size, void* d_ws, size_t ws_size,
                              hipStream_t stream)
{
    (void)in_sizes; (void)n_in; (void)out_size; (void)ws_size;

    const int B_ = 2048, D_IN = 512, E2 = 512, E = 256, NSPK = 50000;

    const float* features = (const float*)d_in[0];   // [2048, 512]
    const int*   labels   = (const int*)d_in[1];     // [2048]
    const float* W1       = (const float*)d_in[2];   // [512, 512]
    const float* b1       = (const float*)d_in[3];   // [512]
    const float* W2       = (const float*)d_in[4];   // [256, 512]
    const float* b2       = (const float*)d_in[5];   // [256]
    const float* arc_w    = (const float*)d_in[6];   // [50000, 256]

    float* out      = (float*)d_out;
    float* emb_norm = out;                                   // 2048*256
    float* emb      = out + (size_t)B_ * E;                  // 2048*256
    float* logits   = out + (size_t)2 * B_ * E;              // 2048*50000
    float* probs    = logits + (size_t)B_ * NSPK;            // 2048*50000

    float* h      = (float*)d_ws;                            // 2048*512 floats
    float* wn_inv = h + (size_t)B_ * E2;                     // 50000 floats

    // GEMM1: h = relu(features @ W1^T + b1)   M=2048 N=512 K=512
    gemm_nt_wmma<0><<<dim3(E2 / 64, B_ / 128), 256, 0, stream>>>(
        features, W1, b1, h, B_, E2, D_IN, nullptr, nullptr);

    // GEMM2: emb = h @ W2^T + b2              M=2048 N=256 K=512
    gemm_nt_wmma<1><<<dim3(E / 64, B_ / 128), 256, 0, stream>>>(
        h, W2, b2, emb, B_, E, E2, nullptr, nullptr);

    // emb_norm = l2norm(emb)
    rownorm_kernel<<<B_, 256, 0, stream>>>(emb, emb_norm);

    // wn_inv = 1/||arc_weight rows||
    wnorm_kernel<<<(NSPK + 7) / 8, 256, 0, stream>>>(arc_w, wn_inv, NSPK);

    // Arc GEMM + fused margin epilogue -> logits   M=2048 N=50000 K=256
    gemm_nt_wmma<2><<<dim3((NSPK + 63) / 64, B_ / 128), 256, 0, stream>>>(
        emb_norm, arc_w, nullptr, logits, B_, NSPK, E, labels, wn_inv);

    // probs = softmax(logits)
    softmax_kernel<<<B_, 256, 0, stream>>>(logits, probs, NSPK);
}